// ModuleAssociate2_78073915507150
// MI455X (gfx1250) — compile-verified
//
#include <hip/hip_runtime.h>
#include <hip/hip_bf16.h>

// ---------------------------------------------------------------------------
// MI455X (gfx1250) implementation.
//
// Problem sizes (fixed by reference): NUM_WORDS=4096, IN_F=100, HID=30, OUT=2.
// out is 4096*4096*2 f32 = 128 MiB -> the kernel is output-write bound
// (~5.8 us at 23.3 TB/s). Stage 1 (WMMA GEMM + row reductions) is ~25 MFLOP
// and negligible; stage 2 is a pure streaming broadcast-add-relu.
// ---------------------------------------------------------------------------

typedef __attribute__((ext_vector_type(2))) float v2f;
typedef __attribute__((ext_vector_type(4))) float v4f;
typedef __attribute__((ext_vector_type(8))) float v8f;

#define NW   4096
#define IN_F 100
#define HID  30

// ---------------------------------------------------------------------------
// Stage 1: one wave (32 threads) per 16-row tile of oo.
//   acc = oo_tile(16x100) @ Wp(100x30) via V_WMMA_F32_16X16X4_F32,
//   two N-tiles (cols 0..15, 16..29 zero-padded to 32), K in steps of 4.
//   Wp is staged into LDS once, padded to 32 cols and paired along K, so the
//   WMMA loop is branch-free: global_load_b64 (A) + 2x ds_load_b64 (B) + 2 WMMA.
//   Then bias+relu -> LDS, and lanes 0..15 reduce each row against
//   W1 = Wc[0:30,:] and W2 = Wc[30:60,:] producing S1[row,2], S2[row,2].
// ---------------------------------------------------------------------------
__global__ __launch_bounds__(32)
void stage1_wmma(const float* __restrict__ oo, const float* __restrict__ Wp,
                 const float* __restrict__ bp, const float* __restrict__ Wc,
                 float* __restrict__ S1, float* __restrict__ S2) {
  const int lane = threadIdx.x;        // 0..31, wave32
  const int half = lane >> 4;          // 0: lanes 0-15, 1: lanes 16-31
  const int l15  = lane & 15;
  const int row0 = blockIdx.x * 16;    // 256 blocks cover 4096 rows

  // Bs[kk][n] = { Wp[2kk, n], Wp[2kk+1, n] }, cols 30/31 zero-padded.
  __shared__ v2f Bs[IN_F / 2][32];     // 50 x 32 x 8B = 12.8 KB
  if (lane < HID) {
#pragma unroll 5
    for (int kk = 0; kk < IN_F / 2; ++kk) {
      v2f p;
      p.x = Wp[(2 * kk) * HID + lane];
      p.y = Wp[(2 * kk + 1) * HID + lane];
      Bs[kk][lane] = p;
    }
  } else {
#pragma unroll 5
    for (int kk = 0; kk < IN_F / 2; ++kk) {
      Bs[kk][lane] = (v2f){0.0f, 0.0f};
    }
  }
  __syncthreads();

  v8f acc0 = {};                       // cols 0..15
  v8f acc1 = {};                       // cols 16..31 (30,31 zero-padded)

  // A fragment (16x4 f32): VGPR0 holds K = k0 + 2*half, VGPR1 holds K+1.
  // Row stride is 400 B and kA is even -> v2f load is 8-byte aligned.
  const float* __restrict__ arow = oo + (size_t)(row0 + l15) * IN_F;

  for (int k0 = 0; k0 < IN_F; k0 += 4) {       // 100 = 25 * 4, exact
    const int kA = k0 + 2 * half;
    const v2f a  = *reinterpret_cast<const v2f*>(arow + kA);
    const v2f b0 = Bs[kA >> 1][l15];           // cols 0..15
    const v2f b1 = Bs[kA >> 1][16 + l15];      // cols 16..31

    acc0 = __builtin_amdgcn_wmma_f32_16x16x4_f32(false, a, false, b0,
                                                 (short)0, acc0, false, false);
    acc1 = __builtin_amdgcn_wmma_f32_16x16x4_f32(false, a, false, b1,
                                                 (short)0, acc1, false, false);
  }

  // C/D layout: VGPR r, lanes 0-15 -> M=r, lanes 16-31 -> M=r+8; N=lane&15.
  __shared__ float Atile[16][32];
  const int  n1  = 16 + l15;
  const bool n1v = (n1 < HID);
  const float bp0 = bp[l15];
  const float bp1 = n1v ? bp[n1] : 0.0f;       // out of WMMA loop; cheap
#pragma unroll
  for (int r = 0; r < 8; ++r) {
    const int row = r + 8 * half;      // local row 0..15
    float v0 = acc0[r] + bp0;
    float v1 = n1v ? (acc1[r] + bp1) : 0.0f;
    Atile[row][l15]      = v0 > 0.0f ? v0 : 0.0f;
    Atile[row][16 + l15] = v1 > 0.0f ? v1 : 0.0f;
  }
  __syncthreads();

  if (lane < 16) {
    float s10 = 0.0f, s11 = 0.0f, s20 = 0.0f, s21 = 0.0f;
#pragma unroll
    for (int c = 0; c < HID; ++c) {
      const float a = Atile[lane][c];
      s10 += a * Wc[c * 2 + 0];              // W1 = Wc[0:30]
      s11 += a * Wc[c * 2 + 1];
      s20 += a * Wc[(HID + c) * 2 + 0];      // W2 = Wc[30:60]
      s21 += a * Wc[(HID + c) * 2 + 1];
    }
    const int row = row0 + lane;
    S1[row * 2 + 0] = s10;
    S1[row * 2 + 1] = s11;
    S2[row * 2 + 0] = s20;
    S2[row * 2 + 1] = s21;
  }
}

// ---------------------------------------------------------------------------
// Stage 2: out[(i*4096 + j)*2 + c] = relu(S1[i,c] + S2[j,c] + bc[c]).
// i is uniform per block (blockIdx.y) -> S1/bc become scalar loads.
// Each lane writes coalesced 16-byte non-temporal stores (2 j's per store),
// 4 stores per thread. One output row = 2048 v4f's = 2 blocks of 256 thr.
// ---------------------------------------------------------------------------
__global__ __launch_bounds__(256)
void stage2_outer(const float* __restrict__ S1, const float* __restrict__ S2,
                  const float* __restrict__ bc, float* __restrict__ out) {
  const int i = blockIdx.y;
  const float p0 = S1[i * 2 + 0] + bc[0];
  const float p1 = S1[i * 2 + 1] + bc[1];

  v4f* __restrict__ orow = reinterpret_cast<v4f*>(out) + (size_t)i * (NW / 2);
  const v4f* __restrict__ q4 = reinterpret_cast<const v4f*>(S2);

  const int base = blockIdx.x * (256 * 4);   // v4f index within the row
#pragma unroll
  for (int u = 0; u < 4; ++u) {
    const int f4 = base + u * 256 + threadIdx.x;   // 0..2047
    const v4f q = q4[f4];                          // S2[j..j+1, 0..1], j = 2*f4
    v4f o;
    o.x = fmaxf(p0 + q.x, 0.0f);
    o.y = fmaxf(p1 + q.y, 0.0f);
    o.z = fmaxf(p0 + q.z, 0.0f);
    o.w = fmaxf(p1 + q.w, 0.0f);
    __builtin_nontemporal_store(o, orow + f4);     // streaming 128 MiB write
  }
}

extern "C" void kernel_launch(void* const* d_in, const int* in_sizes, int n_in,
                              void* d_out, int out_size, void* d_ws, size_t ws_size,
                              hipStream_t stream) {
  const float* oo = (const float*)d_in[0];   // [4096,100]
  const float* Wp = (const float*)d_in[1];   // [100,30]
  const float* bp = (const float*)d_in[2];   // [30]
  const float* Wc = (const float*)d_in[3];   // [60,2]
  const float* bc = (const float*)d_in[4];   // [2]
  float* out = (float*)d_out;                // [4096*4096, 2]

  float* S1 = (float*)d_ws;                  // 4096*2 f32 = 32 KB
  float* S2 = S1 + (size_t)NW * 2;           // 4096*2 f32 = 32 KB

  stage1_wmma<<<dim3(NW / 16), dim3(32), 0, stream>>>(oo, Wp, bp, Wc, S1, S2);

  // Row has 4096*2 floats = 2048 v4f; 256 threads * 4 stores = 1024/block.
  stage2_outer<<<dim3(2, NW), dim3(256), 0, stream>>>(S1, S2, bc, out);
}